// MCPBRNN_SW_Variant_Routing_Norm_10196252360857
// MI455X (gfx1250) — compile-verified
//
#include <hip/hip_runtime.h>
#include <cstdint>
#include <cstddef>

// Problem constants from the reference (B=64, T=4096, H=1).
#define Bsz  64
#define Tlen 4096

typedef int v4i __attribute__((vector_size(16)));               // matches builtin param type
typedef __attribute__((address_space(1))) v4i* gv4_p;           // global int4*
typedef __attribute__((address_space(3))) v4i* lv4_p;           // LDS int4*
typedef __attribute__((address_space(1))) void GAS;             // global void
typedef __attribute__((address_space(3))) void LAS;             // LDS void

// Async DMA of 16B/lane (512B/wave32) from global to LDS. Tracked by ASYNCcnt.
__device__ __forceinline__ void async_b128(const float* g, float* l) {
#if __has_builtin(__builtin_amdgcn_global_load_async_to_lds_b128)
  __builtin_amdgcn_global_load_async_to_lds_b128(
      (gv4_p)(GAS*)g, (lv4_p)(LAS*)l, /*offset=*/0, /*cpol=*/0);
#else
  unsigned int loff = (unsigned int)(unsigned long long)(LAS*)l;  // LDS byte offset
  asm volatile("global_load_async_to_lds_b128 %0, %1, off"
               :: "v"(loff), "v"(g) : "memory");
#endif
}

#if __has_builtin(__builtin_amdgcn_s_wait_asynccnt)
#define WAIT_ASYNC(n) __builtin_amdgcn_s_wait_asynccnt(n)
#else
#define WAIT_ASYNC(n) asm volatile("s_wait_asynccnt %0" :: "i"(n) : "memory")
#endif

#if __has_builtin(__builtin_amdgcn_tanhf)
#define HAVE_HW_TANH 1
#else
#define HAVE_HW_TANH 0
#endif

// oo = oo1 * sigmoid(b0 + c*w/p).  With HW tanh: sigmoid(z) = 0.5*tanh(z/2)+0.5,
// so with kA = 0.5*w/p, kB = 0.5*b0, kC = 0.5*oo1:  oo = kC*tanh(kA*c+kB) + kC.
__device__ __forceinline__ float oo_of_c(float c, float kA, float kB, float kC) {
#if HAVE_HW_TANH
  return fmaf(__builtin_amdgcn_tanhf(fmaf(c, kA, kB)), kC, kC);
#else
  float e = __expf(fmaf(c, kA, kB));       // kA = -w/p, kB = -b0, kC = oo1
  return __fdividef(kC, 1.0f + e);
#endif
}

__global__ void __launch_bounds__(32)
mcpbrnn_scan_kernel(const float* __restrict__ x,
                    const int*   __restrict__ lag_p,
                    const float* __restrict__ p_norm,
                    const float* __restrict__ w_r_yom,
                    const float* __restrict__ w_r_yfm,
                    const float* __restrict__ b0_yom,
                    const float* __restrict__ w_b1_yom,
                    float*       __restrict__ out)
{
  __shared__ float buf[2][Tlen];          // 32 KB double buffer (LDS)
  const int tid = threadIdx.x;

  // Harness poisons d_out: zero all 4*B outputs (covers the time_lag pad rows).
  for (int i = tid; i < 4 * Bsz; i += 32) out[i] = 0.0f;

  const int   lag = lag_p[0];
  const int   nch = Bsz - lag;            // sequential chunks (= rows of x[lag:])
  const float p   = p_norm[0];
  const float eo  = __expf(w_r_yom[0]);
  const float ef  = __expf(w_r_yfm[0]);
  const float oo1 = eo / (eo + ef);       // constant routing ratio

#if HAVE_HW_TANH
  const float kA = 0.5f * (w_b1_yom[0] / p);  // z/2 = kA*c + kB
  const float kB = 0.5f * b0_yom[0];
  const float kC = 0.5f * oo1;                // oo = kC*tanh + kC
#else
  const float kA = -w_b1_yom[0] / p;          // exp(kA*c + kB) = exp(-z)
  const float kB = -b0_yom[0];
  const float kC = oo1;
#endif

  const float* xbase = x + (size_t)lag * Tlen;

  // Prime the pipeline: async-DMA chunk 0 into LDS buffer 0.
  {
    const float* g = xbase;
    float*       l = &buf[0][0];
#pragma unroll
    for (int i = 0; i < 32; ++i)
      async_b128(g + i * 128 + tid * 4, l + i * 128 + tid * 4);
  }

  float c = 0.0f;                         // carry persists across rows (as in ref)

  for (int k = 0; k < nch; ++k) {
    if (k + 1 < nch) {
      // Issue DMA for the next row, then wait for the current row to land.
      const float* g = xbase + (size_t)(k + 1) * Tlen;
      float*       l = &buf[(k + 1) & 1][0];
#pragma unroll
      for (int i = 0; i < 32; ++i)
        async_b128(g + i * 128 + tid * 4, l + i * 128 + tid * 4);
      WAIT_ASYNC(32);                     // in-order ⇒ chunk k fully in LDS
    } else {
      WAIT_ASYNC(0);
    }

    const float* u = &buf[k & 1][0];

#pragma unroll 8
    for (int t = 0; t < Tlen - 1; ++t) {
      float uu = u[t];
#if HAVE_HW_TANH
      // c' = (1-oo)*c + u  with  oo = kC*th + kC  rewritten as
      // c' = s' - m*th,  m = kC*c,  s' = (c+u) - m   (m, s' off critical path).
      // Critical chain: fma -> v_tanh_f32 -> fma   (3 dependent ops).
      float t0 = fmaf(c, kA, kB);
      float m  = kC * c;
      float s2 = (c + uu) - m;
      float th = __builtin_amdgcn_tanhf(t0);
      c = fmaf(-m, th, s2);
#else
      float s  = c + uu;
      float oo = oo_of_c(c, kA, kB, kC);
      c = fmaf(-oo, c, s);
#endif
    }

    // Last timestep of the row: record (oo*c, c, oo, 1-oo) from PRE-update c.
    {
      float uu = u[Tlen - 1];
      float s  = c + uu;
      float oo = oo_of_c(c, kA, kB, kC);
      if (tid == 0) {
        int r = lag + k;
        out[r]           = oo * c;    // h_n record = oo * c_old
        out[Bsz + r]     = c;         // c_n record = c_old
        out[2 * Bsz + r] = oo;        // g_oo
        out[3 * Bsz + r] = 1.0f - oo; // g_f
      }
      c = fmaf(-oo, c, s);
    }
  }
}

extern "C" void kernel_launch(void* const* d_in, const int* in_sizes, int n_in,
                              void* d_out, int out_size, void* d_ws, size_t ws_size,
                              hipStream_t stream) {
  // setup_inputs order: x, epoch, time_lag, y_obs, p_norm, w_r_yom, w_r_yfm,
  //                     b0_yom, w_b1_yom
  const float* x        = (const float*)d_in[0];
  const int*   time_lag = (const int*)  d_in[2];
  const float* p_norm   = (const float*)d_in[4];
  const float* w_r_yom  = (const float*)d_in[5];
  const float* w_r_yfm  = (const float*)d_in[6];
  const float* b0_yom   = (const float*)d_in[7];
  const float* w_b1_yom = (const float*)d_in[8];

  // One wave32: the recurrence is a single dependent chain; async global->LDS
  // double-buffering hides all memory latency behind the VALU chain.
  mcpbrnn_scan_kernel<<<1, 32, 0, stream>>>(x, time_lag, p_norm, w_r_yom,
                                            w_r_yfm, b0_yom, w_b1_yom,
                                            (float*)d_out);
}